// LightGCN_48258252538258
// MI455X (gfx1250) — compile-verified
//
#include <hip/hip_runtime.h>

// ---------------- problem constants (match reference) ----------------
#define NUM_USERS 100000
#define NUM_ITEMS 200000
#define N_TOTAL   300000
#define EMBED_DIM 64
#define NNZ       4000000
#define BATCH     16384

typedef __attribute__((ext_vector_type(2))) float v2f;
typedef __attribute__((ext_vector_type(4))) float v4f;
typedef __attribute__((ext_vector_type(8))) float v8f;

// ---------------------------------------------------------------------
// Layer-0 contribution: u_acc[b] = user_emb[uid[b]], i_acc[b] = item_emb[iid[b]]
// 1 thread per (b, d) element.
__global__ void lgcn_init_acc(const int* __restrict__ uid, const int* __restrict__ iid,
                              const float* __restrict__ uw, const float* __restrict__ iw,
                              float* __restrict__ uacc, float* __restrict__ iacc) {
    int t = blockIdx.x * blockDim.x + threadIdx.x;     // BATCH*64 threads
    int b = t >> 6, d = t & 63;
    if (b >= BATCH) return;
    uacc[t] = uw[(long)uid[b] * EMBED_DIM + d];
    iacc[t] = iw[(long)iid[b] * EMBED_DIM + d];
}

// ---------------------------------------------------------------------
// COO SpMM: y[row] += val * x[col].  16 lanes per edge, float4 per lane
// (16*4 = 64 = D): gather of one source row is a fully-coalesced 256B read.
// L2-residency strategy: the two 76.8MB layer buffers (153.6MB) live in the
// 192MB L2 with RT hints (rows reused ~13x on average), while the 48MB/layer
// of streaming edge metadata is loaded NON-TEMPORAL so it doesn't evict them.
// Scatter = 4 hardware global_atomic_add_f32 per lane (relaxed, agent scope,
// no return -> STOREcnt-tracked, no CAS loop).
// split=1: layer-1 reads directly from the two embedding tables (avoids
// materializing the concatenated x0).
__global__ void lgcn_spmm(const int* __restrict__ rowi, const int* __restrict__ coli,
                          const float* __restrict__ vali,
                          const float* __restrict__ xu, const float* __restrict__ xi,
                          int split, float* __restrict__ y) {
    long t = (long)blockIdx.x * blockDim.x + threadIdx.x;
    long e = t >> 4;
    int sub = (int)(t & 15);
    if (e >= NNZ) return;
    // streaming, read-once-per-layer: non-temporal so L2 keeps the embeddings
    int   r = __builtin_nontemporal_load(rowi + e);
    int   c = __builtin_nontemporal_load(coli + e);
    float v = __builtin_nontemporal_load(vali + e);
    const float* src;
    if (split) {
        src = (c < NUM_USERS) ? (xu + (long)c * EMBED_DIM)
                              : (xi + (long)(c - NUM_USERS) * EMBED_DIM);
    } else {
        src = xu + (long)c * EMBED_DIM;
    }
    v4f x = *(const v4f*)(src + sub * 4);              // RT: L2-resident gather
    float* dst = y + (long)r * EMBED_DIM + sub * 4;
    __hip_atomic_fetch_add(dst + 0, v * x.x, __ATOMIC_RELAXED, __HIP_MEMORY_SCOPE_AGENT);
    __hip_atomic_fetch_add(dst + 1, v * x.y, __ATOMIC_RELAXED, __HIP_MEMORY_SCOPE_AGENT);
    __hip_atomic_fetch_add(dst + 2, v * x.z, __ATOMIC_RELAXED, __HIP_MEMORY_SCOPE_AGENT);
    __hip_atomic_fetch_add(dst + 3, v * x.w, __ATOMIC_RELAXED, __HIP_MEMORY_SCOPE_AGENT);
}

// ---------------------------------------------------------------------
// After each layer: accumulate the batch rows of y into u_acc / i_acc.
__global__ void lgcn_gather_acc(const int* __restrict__ uid, const int* __restrict__ iid,
                                const float* __restrict__ ybuf,
                                float* __restrict__ uacc, float* __restrict__ iacc) {
    int t = blockIdx.x * blockDim.x + threadIdx.x;     // BATCH*64 threads
    int b = t >> 6, d = t & 63;
    if (b >= BATCH) return;
    uacc[t] += ybuf[(long)uid[b] * EMBED_DIM + d];
    iacc[t] += ybuf[((long)iid[b] + NUM_USERS) * EMBED_DIM + d];
}

// ---------------------------------------------------------------------
// Readout via the matrix pipe: one wave32 handles 16 (user,item) pairs.
// D = U * I^T with V_WMMA_F32_16X16X4_F32, K=64 accumulated as 16 chunks
// of 4 in full fp32; the scores are the diagonal of D.
// A 16x4 f32 layout (ISA 7.12.2): lane l -> M = l&15, K = 2*(l>>4)+{0,1}.
// B 4x16 f32 layout is row-striped across lanes symmetrically, so the
// per-lane fetch address is identical for both fragments.
// C/D 16x16 f32 layout: VGPR r, lanes 0-15 -> (M=r, N=lane);
//                       lanes 16-31 -> (M=r+8, N=lane-16).
// Diagonal: M<8 at (vgpr=M, lane=M); M>=8 at (vgpr=M-8, lane=M+16).
__global__ void lgcn_score_wmma(const float* __restrict__ uacc,
                                const float* __restrict__ iacc,
                                float* __restrict__ out) {
    int wave = (blockIdx.x * blockDim.x + threadIdx.x) >> 5;
    int lane = threadIdx.x & 31;
    int base = wave * 16;                 // 1024 waves * 16 = BATCH exactly
    if (base >= BATCH) return;            // wave-uniform: EXEC stays all-ones

    int m    = lane & 15;
    int koff = (lane >> 4) * 2;
    long ridx = (long)(base + m) * EMBED_DIM + koff;

    v8f c = {0.f, 0.f, 0.f, 0.f, 0.f, 0.f, 0.f, 0.f};
#pragma unroll
    for (int k0 = 0; k0 < EMBED_DIM; k0 += 4) {
        v2f a = *(const v2f*)(uacc + ridx + k0);   // A[m][k0+koff .. +1]
        v2f b = *(const v2f*)(iacc + ridx + k0);   // B[k0+koff][n=m] row-pair
        c = __builtin_amdgcn_wmma_f32_16x16x4_f32(
                /*neg_a=*/false, a, /*neg_b=*/false, b,
                /*c_mod=*/(short)0, c, /*reuse_a=*/false, /*reuse_b=*/false);
    }

    // extract diagonal of D
    float s = 0.f;
#pragma unroll
    for (int r = 0; r < 8; ++r) {
        if (lane == r)      s = c[r];   // (M=r,  N=r)
        if (lane == r + 24) s = c[r];   // (M=r+8,N=r+8)
    }
    // final = acc/4 for both sides -> dot scales by 1/16
    if (lane < 8)        out[base + lane]      = s * 0.0625f;
    else if (lane >= 24) out[base + lane - 16] = s * 0.0625f;
}

// ---------------------------------------------------------------------
extern "C" void kernel_launch(void* const* d_in, const int* in_sizes, int n_in,
                              void* d_out, int out_size, void* d_ws, size_t ws_size,
                              hipStream_t stream) {
    const int*   user_ids = (const int*)  d_in[0];
    const int*   item_ids = (const int*)  d_in[1];
    const int*   adj_row  = (const int*)  d_in[2];
    const int*   adj_col  = (const int*)  d_in[3];
    const float* adj_val  = (const float*)d_in[4];
    const float* uw       = (const float*)d_in[5];
    const float* iw       = (const float*)d_in[6];
    float*       out      = (float*)d_out;

    // workspace layout: two L2-resident layer buffers + batch accumulators
    char*  ws       = (char*)d_ws;
    size_t bufBytes = (size_t)N_TOTAL * EMBED_DIM * sizeof(float);   // 76.8 MB
    size_t accBytes = (size_t)BATCH  * EMBED_DIM * sizeof(float);    // 4 MB
    float* buf0 = (float*)(ws);
    float* buf1 = (float*)(ws + bufBytes);
    float* uacc = (float*)(ws + 2 * bufBytes);
    float* iacc = (float*)(ws + 2 * bufBytes + accBytes);

    const int TB = 256;
    int accGrid  = (BATCH * EMBED_DIM) / TB;          // 4096
    int spmmGrid = (int)(((long)NNZ * 16) / TB);      // 250000

    // layer 0 contribution + zero scatter target
    hipMemsetAsync(buf0, 0, bufBytes, stream);
    lgcn_init_acc<<<accGrid, TB, 0, stream>>>(user_ids, item_ids, uw, iw, uacc, iacc);

    // layer 1: (split tables) -> buf0
    lgcn_spmm<<<spmmGrid, TB, 0, stream>>>(adj_row, adj_col, adj_val, uw, iw, 1, buf0);
    hipMemsetAsync(buf1, 0, bufBytes, stream);
    lgcn_gather_acc<<<accGrid, TB, 0, stream>>>(user_ids, item_ids, buf0, uacc, iacc);

    // layer 2: buf0 -> buf1
    lgcn_spmm<<<spmmGrid, TB, 0, stream>>>(adj_row, adj_col, adj_val, buf0, nullptr, 0, buf1);
    lgcn_gather_acc<<<accGrid, TB, 0, stream>>>(user_ids, item_ids, buf1, uacc, iacc);
    hipMemsetAsync(buf0, 0, bufBytes, stream);       // safe: after spmm2 read of buf0

    // layer 3: buf1 -> buf0
    lgcn_spmm<<<spmmGrid, TB, 0, stream>>>(adj_row, adj_col, adj_val, buf1, nullptr, 0, buf0);
    lgcn_gather_acc<<<accGrid, TB, 0, stream>>>(user_ids, item_ids, buf0, uacc, iacc);

    // WMMA readout: 1024 waves of 32, 4 waves per 128-thread block -> 256 blocks
    int scoreGrid = (BATCH / 16) * 32 / 128;
    lgcn_score_wmma<<<scoreGrid, 128, 0, stream>>>(uacc, iacc, out);
}